// SimGCN_8641474200133
// MI455X (gfx1250) — compile-verified
//
#include <hip/hip_runtime.h>
#include <hip/hip_bf16.h>
#include <stdint.h>

#define N_USER 100000
#define N_ITEM 200000
#define N_TOTAL (N_USER + N_ITEM)
#define NNZ 9600000
#define DIM 64

typedef __attribute__((ext_vector_type(4))) unsigned int v4u;
typedef __attribute__((ext_vector_type(8))) int          v8i;
typedef __attribute__((ext_vector_type(4))) int          v4i;

// ---------------------------------------------------------------------------
// TDM gather-mode descriptor (CDNA5 ISA ch.8): fetch nIdx rows of 64 f32
// (256B each) from `table` (tableRows x 64, row-major) into LDS at ldsByteOff,
// rows packed consecutively. 32-bit index mode => up to 8 rows per descriptor.
// clang-23 builtin arity is 6: (g0 v4u, g1 v8i, g2 v4i, g3 v4i, v8i, i32 cpol);
// indices are supplied both in D# groups 2/3 (ISA layout) and the v8i operand.
// ---------------------------------------------------------------------------
__device__ __forceinline__ void tdm_gather_rows(const float* table,
                                                unsigned tableRows,
                                                unsigned ldsByteOff,
                                                const int* idx, int nIdx) {
  unsigned long long ga = (unsigned long long)(uintptr_t)table;

  // Group 0: count=1, gather_index_size=32b, gather_mode=1, lds_addr,
  // global_addr[56:0], type=2 ("image")
  v4u g0;
  g0.x = 1u | (1u << 30) | (1u << 31);
  g0.y = ldsByteOff;
  g0.z = (unsigned)(ga & 0xFFFFFFFFu);
  g0.w = (unsigned)((ga >> 32) & 0x01FFFFFFu) | (2u << 30);

  // Group 1: data_size=4B(code 2), tensor_dim0=64, tensor_dim1=tableRows,
  // tile_dim0=64, tile_dim1=#indices, tensor_dim0_stride=64
  v8i g1;
  g1[0] = (int)(2u << 16);                                  // data_size
  g1[1] = (int)(64u << 16);                                 // tensor_dim0[15:0]
  g1[2] = (int)((tableRows & 0xFFFFu) << 16);               // dim0 hi | dim1 lo
  g1[3] = (int)((tableRows >> 16) | (64u << 16));           // dim1 hi | tile_dim0
  g1[4] = nIdx;                                             // tile_dim1 (=#indices)
  g1[5] = 64;                                               // tensor_dim0_stride lo
  g1[6] = 0;
  g1[7] = 0;

  // Groups 2/3: 32-bit row indices 0..7 (ISA D# gather layout)
  v4i g2 = {idx[0], idx[1], idx[2], idx[3]};
  v4i g3 = {idx[4], idx[5], idx[6], idx[7]};
  // clang-23's extra v8i operand: same 8 gather indices.
  v8i gi = {idx[0], idx[1], idx[2], idx[3], idx[4], idx[5], idx[6], idx[7]};

  __builtin_amdgcn_tensor_load_to_lds(g0, g1, g2, g3, gi, 0);
}

// ---------------------------------------------------------------------------
// Kernel 1: out[n] = emb[n] + sum_k w[n,k] * table[nbr[n,k]]
// One wave32 per output row; TDM gathers the 10 neighbor rows into LDS.
// ---------------------------------------------------------------------------
__global__ void __launch_bounds__(256) simgcn_neighbor_kernel(
    const float* __restrict__ users_emb, const float* __restrict__ items_emb,
    const int* __restrict__ u_nbr, const float* __restrict__ u_w,
    const int* __restrict__ i_nbr, const float* __restrict__ i_w,
    float* __restrict__ out) {
  __shared__ float lds[8 * 10 * DIM];  // 8 waves * 10 rows * 64 f32 = 20 KB

  const int lane = threadIdx.x & 31;
  const int wave = threadIdx.x >> 5;
  const long long n = (long long)blockIdx.x * 8 + wave;  // 300000 % 8 == 0
  if (n >= N_TOTAL) return;

  const float* table;
  const float* self;
  const int* nbr;
  const float* wgt;
  unsigned rows;
  if (n < N_USER) {
    table = users_emb;           rows = N_USER;
    nbr = u_nbr + n * 10;        wgt = u_w + n * 10;
    self = users_emb + n * DIM;
  } else {
    long long m = n - N_USER;
    table = items_emb;           rows = N_ITEM;
    nbr = i_nbr + m * 10;        wgt = i_w + m * 10;
    self = items_emb + m * DIM;
  }

  // Coalesced load of the 10 wave-uniform neighbor ids, hoist to SGPRs.
  int myidx = 0;
  if (lane < 10) myidx = __builtin_nontemporal_load(nbr + lane);
  int idx[10];
#pragma unroll
  for (int k = 0; k < 10; ++k) idx[k] = __builtin_amdgcn_readlane(myidx, k);

  const unsigned ldsOff = (unsigned)(uintptr_t)&lds[wave * 10 * DIM];
  tdm_gather_rows(table, rows, ldsOff, idx, 8);            // rows 0..7
  int idx2[8] = {idx[8], idx[9], 0, 0, 0, 0, 0, 0};
  tdm_gather_rows(table, rows, ldsOff + 8 * 256, idx2, 2); // rows 8..9

  // Overlap: self row load while TDM runs. 2 f32 per lane covers 64 cols.
  float2 acc = *(const float2*)(self + lane * 2);

  __builtin_amdgcn_s_wait_tensorcnt(0);

  const float* L = &lds[wave * 10 * DIM];
#pragma unroll
  for (int k = 0; k < 10; ++k) {
    float w = __builtin_nontemporal_load(wgt + k);         // wave-broadcast
    float2 v = *(const float2*)(L + k * DIM + lane * 2);   // conflict-free b64
    acc.x += w * v.x;
    acc.y += w * v.y;
  }
  *(float2*)(out + n * DIM + lane * 2) = acc;
}

// ---------------------------------------------------------------------------
// Kernel 2: out[rows[e]] += vals[e] * emb[cols[e]]   (segment_sum via L2 atomics)
// 16 threads per edge, float4 each: 256B coalesced gather + 4 f32 atomics/lane.
// emb (76.8MB) + out (76.8MB) both fit in the 192MB L2 -> atomics resolve in L2.
// ---------------------------------------------------------------------------
__global__ void __launch_bounds__(256) simgcn_spmm_kernel(
    const float* __restrict__ users_emb, const float* __restrict__ items_emb,
    const int* __restrict__ rows, const int* __restrict__ cols,
    const float* __restrict__ vals, float* __restrict__ out) {
  const long long tid = (long long)blockIdx.x * blockDim.x + threadIdx.x;
  const long long e = tid >> 4;
  const int c = (int)(tid & 15);
  if (e >= NNZ) return;

  const int r   = __builtin_nontemporal_load(rows + e);  // streams: keep out of L2
  const int col = __builtin_nontemporal_load(cols + e);
  const float v = __builtin_nontemporal_load(vals + e);

  const float* src = (col < N_USER)
                         ? (users_emb + (long long)col * DIM)
                         : (items_emb + (long long)(col - N_USER) * DIM);
  const float4 x = *(const float4*)(src + c * 4);  // RT: want emb resident in L2

  float* dst = out + (long long)r * DIM + c * 4;
  unsafeAtomicAdd(dst + 0, v * x.x);  // global_atomic_add_f32, no return
  unsafeAtomicAdd(dst + 1, v * x.y);
  unsafeAtomicAdd(dst + 2, v * x.z);
  unsafeAtomicAdd(dst + 3, v * x.w);
}

extern "C" void kernel_launch(void* const* d_in, const int* in_sizes, int n_in,
                              void* d_out, int out_size, void* d_ws, size_t ws_size,
                              hipStream_t stream) {
  const float* users_emb = (const float*)d_in[0];
  const float* items_emb = (const float*)d_in[1];
  const int*   u_nbr     = (const int*)d_in[2];
  const float* u_w       = (const float*)d_in[3];
  const int*   i_nbr     = (const int*)d_in[4];
  const float* i_w       = (const float*)d_in[5];
  const int*   g_rows    = (const int*)d_in[6];
  const int*   g_cols    = (const int*)d_in[7];
  const float* g_vals    = (const float*)d_in[8];
  float* out = (float*)d_out;

  // Kernel 1: fills every output row with self + neighbor aggregation.
  const int blocks1 = (N_TOTAL + 7) / 8;  // 8 waves (rows) per 256-thread block
  simgcn_neighbor_kernel<<<blocks1, 256, 0, stream>>>(
      users_emb, items_emb, u_nbr, u_w, i_nbr, i_w, out);

  // Kernel 2: edge-parallel scatter-add on top.
  const long long threads2 = (long long)NNZ * 16;
  const int blocks2 = (int)((threads2 + 255) / 256);
  simgcn_spmm_kernel<<<blocks2, 256, 0, stream>>>(
      users_emb, items_emb, g_rows, g_cols, g_vals, out);
}